// PQLayer_52733608460671
// MI455X (gfx1250) — compile-verified
//
#include <hip/hip_runtime.h>

typedef __attribute__((ext_vector_type(2))) float v2f;
typedef __attribute__((ext_vector_type(4))) float v4f;
typedef __attribute__((ext_vector_type(8))) float v8f;

#define FEAT  512
#define MSUB  64
#define KCW   256
#define DSUB  8
#define SCALE 64.0f   // M / ALPHA

// One wave = one subspace m x 16 batch rows.
// ips tile (16 x 256) via 16 N-tiles of V_WMMA_F32_16X16X4_F32 (2 K-steps for D=8).
__global__ __launch_bounds__(256) void pq_fused_kernel(
    const float* __restrict__ x,      // (B, 512)
    const float* __restrict__ Cb,     // (64, 256, 8)
    float* __restrict__ xhat,         // (B, 512)
    float* __restrict__ codes)        // (B, 64, 256)
{
    const int lane  = threadIdx.x & 31;
    const int wave  = threadIdx.x >> 5;
    const int m     = blockIdx.y * 8 + wave;      // subspace
    const int bbase = blockIdx.x * 16;            // batch tile

    const int ln16  = lane & 15;                  // row/col within 16-group
    const int kg    = (lane >> 4) << 1;           // K pair base: 0 or 2

    // ---------------------------------------------------------------
    // Phase 0: stream zeros into this wave's codes tile (16 rows x 1KB)
    // with non-temporal b128 stores; drains while WMMA runs below.
    // ---------------------------------------------------------------
    {
        const size_t row0 = ((size_t)bbase * MSUB + m) * KCW;   // float index
        v4f z = {0.f, 0.f, 0.f, 0.f};
#pragma unroll
        for (int i = 0; i < 32; ++i) {
            int c  = i * 32 + lane;          // 0..1023 float4 chunks
            int r  = c >> 6;                 // row 0..15
            int ch = c & 63;                 // float4 chunk within row
            float* p = codes + row0 + (size_t)r * (MSUB * KCW) + ch * 4;
            __builtin_nontemporal_store(z, (v4f*)p);
        }
    }

    // ---------------------------------------------------------------
    // Phase 1: ips = X_m (16x8) @ C_m^T (8x256) via f32 WMMA.
    // A (16x4 f32): lane l -> row ln16, K = kg + {0,1}
    // ---------------------------------------------------------------
    const float* xrow = x + (size_t)(bbase + ln16) * FEAT + m * DSUB + kg;
    v2f a0 = *(const v2f*)xrow;          // d = kg, kg+1       (K-step 0: d 0..3)
    v2f a1 = *(const v2f*)(xrow + 4);    // d = 4+kg, 5+kg     (K-step 1: d 4..7)

    v8f acc[16];
#pragma unroll
    for (int t = 0; t < 16; ++t) {
        // B (4x16 f32): lane l -> col ln16, K = kg + {0,1}
        const float* cb = Cb + (((size_t)m * KCW + t * 16 + ln16) * DSUB) + kg;
        v2f b0 = *(const v2f*)cb;
        v2f b1 = *(const v2f*)(cb + 4);
        v8f c0 = {0.f, 0.f, 0.f, 0.f, 0.f, 0.f, 0.f, 0.f};
        c0 = __builtin_amdgcn_wmma_f32_16x16x4_f32(false, a0, false, b0,
                                                   (short)0, c0, false, false);
        acc[t] = __builtin_amdgcn_wmma_f32_16x16x4_f32(false, a1, false, b1,
                                                       (short)0, c0, false, false);
    }

    // ---------------------------------------------------------------
    // Phase 2: per-row argmax + softmax sum. Row j (VGPR slot j) lives on
    // lanes 0-15 (row bbase+j) and lanes 16-31 (row bbase+j+8); butterfly
    // over xor offsets 1,2,4,8 stays inside each 16-lane half.
    // codes[row, kmax] = (1 - s) + s,  x_hat[row,:] = that * C[m,kmax,:].
    // ---------------------------------------------------------------
#pragma unroll
    for (int j = 0; j < 8; ++j) {
        float mval = acc[0][j];
        int   midx = ln16;                       // k = t*16 + ln16, t=0
#pragma unroll
        for (int t = 1; t < 16; ++t) {
            float v = acc[t][j];
            int   k = (t << 4) + ln16;
            if (v > mval) { mval = v; midx = k; }
        }
#pragma unroll
        for (int off = 1; off <= 8; off <<= 1) {
            float ov = __shfl_xor(mval, off, 32);
            int   oi = __shfl_xor(midx, off, 32);
            if (ov > mval || (ov == mval && oi < midx)) { mval = ov; midx = oi; }
        }
        // sum of exp(64*(v - vmax)) over the row (includes exp(0)=1 for max)
        float s = 0.f;
#pragma unroll
        for (int t = 0; t < 16; ++t)
            s += __expf(SCALE * (acc[t][j] - mval));
#pragma unroll
        for (int off = 1; off <= 8; off <<= 1)
            s += __shfl_xor(s, off, 32);

        float soft = 1.0f / s;                   // softmax value at argmax
        float q    = (1.0f - soft) + soft;       // straight-through value (== ref fp math)

        const int row = bbase + j + ((lane >> 4) << 3);

        // scatter the single nonzero of codes (one lane per 16-group)
        if (ln16 == 0)
            codes[((size_t)row * MSUB + m) * KCW + midx] = q;

        // x_hat[row, m*8 + d] = q * C[m, midx, d], d handled by lanes 0..7 of group
        if (ln16 < 8) {
            int d = ln16;
            xhat[(size_t)row * FEAT + m * DSUB + d] =
                q * Cb[((size_t)m * KCW + midx) * DSUB + d];
        }
    }
}

extern "C" void kernel_launch(void* const* d_in, const int* in_sizes, int n_in,
                              void* d_out, int out_size, void* d_ws, size_t ws_size,
                              hipStream_t stream) {
    const float* x  = (const float*)d_in[0];          // (B, 512) f32
    const float* Cb = (const float*)d_in[1];          // (64, 256, 8) f32
    const int B = in_sizes[0] / FEAT;                 // 16384

    float* xhat  = (float*)d_out;                     // first output, B*512
    float* codes = (float*)d_out + (size_t)B * FEAT;  // second output, B*64*256

    dim3 grid(B / 16, MSUB / 8);
    pq_fused_kernel<<<grid, dim3(256), 0, stream>>>(x, Cb, xhat, codes);
}